// ContrastiveVolumeLoss_76544907149334
// MI455X (gfx1250) — compile-verified
//
#include <hip/hip_runtime.h>
#include <hip/hip_bf16.h>

typedef __attribute__((ext_vector_type(2))) float v2f;
typedef __attribute__((ext_vector_type(8))) float v8f;

#define VOLX 64
#define VOLY 128
#define VOLZ 128
#define VELEMS (VOLX * VOLY * VOLZ)   // 1,048,576 voxels
#define NCH 32                        // channels
#define NPTS 2048                     // N
#define M2 (2 * NPTS)                 // 4096 rows in G
#define NTILES (M2 / 16)              // 256 16x16 tiles per dim
#define INV_T 14.285714285714286f     // 1 / 0.07

// ---------------------------------------------------------------------------
// Kernel 1: gather G[p][c] = emb[c * V + flat_index(loc[p])] ; zero rowsum.
// p < NPTS -> (loc0, e0); p >= NPTS -> (loc1, e1).  Writes are coalesced.
// ---------------------------------------------------------------------------
__global__ void gather_init_kernel(const float* __restrict__ e0,
                                   const float* __restrict__ e1,
                                   const float* __restrict__ loc0,
                                   const float* __restrict__ loc1,
                                   float* __restrict__ G,
                                   float* __restrict__ rowsum) {
  int tid = blockIdx.x * blockDim.x + threadIdx.x;   // 0 .. M2*NCH-1
  int p = tid >> 5;
  int c = tid & 31;
  const float* loc = (p < NPTS) ? loc0 : loc1;
  const float* emb = (p < NPTS) ? e0 : e1;
  int q = (p < NPTS) ? p : (p - NPTS);
  int ix = (int)floorf(loc[q * 3 + 0]);
  int iy = (int)floorf(loc[q * 3 + 1]);
  int iz = (int)floorf(loc[q * 3 + 2]);
  int ind = ix * (VOLY * VOLZ) + iy * VOLZ + iz;
  G[p * NCH + c] = emb[c * VELEMS + ind];
  if (tid < M2) rowsum[tid] = 0.0f;
}

// ---------------------------------------------------------------------------
// Kernel 2: fused sim-tile GEMM (fp32 WMMA 16x16x4, K=32 via 8 steps) +
// exp(sim/T) + diagonal-excluded row sums.  Fully straight-line epilogue:
// no conditional stores, diagonal exclusion via float select (v_cndmask).
// Grid: 256 blocks (one 16-row tile each) x 8 waves; wave w covers column
// tiles {w, w+8, ...}.  C/D layout: element (M,N) with M = v + 8*half in
// VGPR v, N = lane&15.  A layout: lane m holds row m, K = 2*half + v.
// For G*G^T the B-fragment uses the identical per-lane formula on col rows.
// ---------------------------------------------------------------------------
__global__ void sim_rowsum_kernel(const float* __restrict__ G,
                                  float* __restrict__ rowsum) {
  const int tr   = blockIdx.x;           // row tile 0..255
  const int wave = threadIdx.x >> 5;     // 0..7
  const int lane = threadIdx.x & 31;
  const int half = lane >> 4;            // 0 or 1
  const int n    = lane & 15;            // column within tile / A row
  const int row0 = tr * 16;
  const int mloc = n - 8 * half;         // v index at which (n == local row)

  // Preload all 8 A fragments (16 VGPRs) — constant across column tiles.
  v2f a[8];
  const float* Arow = G + (row0 + n) * NCH + 2 * half;
#pragma unroll
  for (int s = 0; s < 8; ++s) a[s] = *(const v2f*)(Arow + 4 * s);

  float rs[8];
#pragma unroll
  for (int v = 0; v < 8; ++v) rs[v] = 0.0f;

  for (int tc = wave; tc < NTILES; tc += 8) {
    const int col0 = tc * 16;
    const float* Brow = G + (col0 + n) * NCH + 2 * half;
    v8f acc = {};
#pragma unroll
    for (int s = 0; s < 8; ++s) {
      v2f b = *(const v2f*)(Brow + 4 * s);
      // D = A(16x4) * B(4x16) + C ; fp32 in, fp32 accumulate
      acc = __builtin_amdgcn_wmma_f32_16x16x4_f32(
          /*neg_a=*/false, a[s], /*neg_b=*/false, b,
          /*c_mod=*/(short)0, acc, /*reuse_a=*/false, /*reuse_b=*/false);
    }
    const bool diagTile = (tc == tr);
#pragma unroll
    for (int v = 0; v < 8; ++v) {
      float e = __expf(acc[v] * INV_T);
      // exclude the (r, r) diagonal element via select, no branches
      float keep = (diagTile && v == mloc) ? 0.0f : 1.0f;
      rs[v] = fmaf(e, keep, rs[v]);
    }
  }

  // Reduce each row partial across the 16 lanes of this half-group.
#pragma unroll
  for (int v = 0; v < 8; ++v) {
    float x = rs[v];
    x += __shfl_xor(x, 1, 32);
    x += __shfl_xor(x, 2, 32);
    x += __shfl_xor(x, 4, 32);
    x += __shfl_xor(x, 8, 32);
    rs[v] = x;
  }
  if (n == 0) {  // lanes 0 and 16 write rows (0..7)+8*half of this tile
#pragma unroll
    for (int v = 0; v < 8; ++v)
      atomicAdd(rowsum + row0 + v + 8 * half, rs[v]);
  }
}

// ---------------------------------------------------------------------------
// Kernel 3: recompute sim(i, i+N) = dot(G[i], G[i+N]) directly (2048 x K=32,
// trivial), then loss = -(1/2N) * sum_i [ 2*s_uv/T - log(sumE_i) - log(sumE_{i+N}) ]
// ---------------------------------------------------------------------------
__global__ void loss_kernel(const float* __restrict__ G,
                            const float* __restrict__ rowsum,
                            float* __restrict__ out) {
  __shared__ float red[256];
  float acc = 0.0f;
  for (int i = threadIdx.x; i < NPTS; i += 256) {
    const float* gi = G + (size_t)i * NCH;
    const float* gj = G + (size_t)(i + NPTS) * NCH;
    float dot = 0.0f;
#pragma unroll
    for (int c = 0; c < NCH; ++c) dot = fmaf(gi[c], gj[c], dot);
    float log_euv = dot * INV_T;  // natural log of exp(sim/T)
    acc += 2.0f * log_euv - logf(rowsum[i]) - logf(rowsum[i + NPTS]);
  }
  red[threadIdx.x] = acc;
  __syncthreads();
  for (int s = 128; s > 0; s >>= 1) {
    if (threadIdx.x < s) red[threadIdx.x] += red[threadIdx.x + s];
    __syncthreads();
  }
  if (threadIdx.x == 0) out[0] = -red[0] / (2.0f * NPTS);
}

// ---------------------------------------------------------------------------
extern "C" void kernel_launch(void* const* d_in, const int* in_sizes, int n_in,
                              void* d_out, int out_size, void* d_ws, size_t ws_size,
                              hipStream_t stream) {
  (void)in_sizes; (void)n_in; (void)out_size; (void)ws_size;
  const float* e0 = (const float*)d_in[0];   // (1, 32, 64,128,128) f32
  const float* e1 = (const float*)d_in[1];
  const float* l0 = (const float*)d_in[2];   // (1, 2048, 3) f32
  const float* l1 = (const float*)d_in[3];

  float* G      = (float*)d_ws;              // 4096 x 32 f32   (512 KB)
  float* rowsum = G + (size_t)M2 * NCH;      // 4096 f32

  gather_init_kernel<<<(M2 * NCH) / 256, 256, 0, stream>>>(e0, e1, l0, l1, G, rowsum);
  sim_rowsum_kernel<<<NTILES, 256, 0, stream>>>(G, rowsum);
  loss_kernel<<<1, 256, 0, stream>>>(G, rowsum, (float*)d_out);
}